// GATv2Edge_60533269070350
// MI455X (gfx1250) — compile-verified
//
#include <hip/hip_runtime.h>
#include <hip/hip_bf16.h>
#include <math.h>

#define BT   1024
#define Nn   64
#define Dd   256
#define NHEAD 4
#define DH   64
#define Ee   512
#define LRELU 0.2f
#define EPSI  1e-8f

typedef __attribute__((ext_vector_type(2))) float v2f;
typedef __attribute__((ext_vector_type(8))) float v8f;

// ---------------------------------------------------------------------------
// Kernel 1: Xq = H @ W_lin^T, Xv = H @ W_val^T   (M=BT*Nn, K=Dd, N=Dd)
// M-tile = 32 rows/block. A tile staged into LDS with async global->LDS DMA.
// Wave w owns (rowtile, matrix, col-half): 8 WMMA tiles, K swept in steps of 4.
// ---------------------------------------------------------------------------
#define K1_STRIDE 264   // padded row stride (multiple of 4 floats -> 16B align)

__global__ __launch_bounds__(256)
void k_dual_gemm(const float* __restrict__ H, const float* __restrict__ Wl,
                 const float* __restrict__ Wv, float* __restrict__ Xq,
                 float* __restrict__ Xv) {
  __shared__ float As[32 * K1_STRIDE];     // 33.8 KB
  const int tid = threadIdx.x;
  const int m0  = blockIdx.x * 32;

  // ---- async stage 32x256 A tile: 2048 float4, 8 per thread ----
#pragma unroll
  for (int i = 0; i < 8; ++i) {
    int idx = tid + i * 256;                 // float4 index
    int row = idx >> 6;                      // (idx*4)/256
    int col = (idx & 63) << 2;
    unsigned int lds = (unsigned int)(uintptr_t)&As[row * K1_STRIDE + col];
    unsigned long long ga =
        (unsigned long long)(H + (size_t)(m0 + row) * Dd + col);
    asm volatile("global_load_async_to_lds_b128 %0, %1, off"
                 :: "v"(lds), "v"(ga) : "memory");
  }
  asm volatile("s_wait_asynccnt 0x0" ::: "memory");
  __syncthreads();

  const int wave  = tid >> 5;
  const int lane  = tid & 31;
  const int lrow  = lane & 15;
  const int khalf = (lane >> 4) << 1;        // 0 or 2
  const int r     = (wave >> 2) & 1;         // row tile (16 rows each)
  const int mat   = (wave >> 1) & 1;         // 0 = Xq/W_lin, 1 = Xv/W_val
  const int ch    = wave & 1;                // col-half: coltiles ch*8..ch*8+7
  const float* W  = mat ? Wv : Wl;

  v8f acc[8] = {};

  for (int k0 = 0; k0 < Dd; k0 += 4) {
    v2f a;
    {
      const float* ap = &As[(r * 16 + lrow) * K1_STRIDE + k0 + khalf];
      a.x = ap[0]; a.y = ap[1];
    }
#pragma unroll
    for (int j = 0; j < 8; ++j) {
      int col = ((ch * 8 + j) << 4) + lrow;  // output column
      const float* bp = W + (size_t)col * Dd + k0 + khalf;
      v2f b; b.x = bp[0]; b.y = bp[1];
      acc[j] = __builtin_amdgcn_wmma_f32_16x16x4_f32(false, a, false, b,
                                                     (short)0, acc[j], false, false);
    }
  }

  float* O = mat ? Xv : Xq;
  const int rbase = (lane >> 4) << 3;        // 0 or 8
#pragma unroll
  for (int j = 0; j < 8; ++j) {
    int n0 = (ch * 8 + j) << 4;
#pragma unroll
    for (int v = 0; v < 8; ++v)
      O[(size_t)(m0 + r * 16 + rbase + v) * Dd + n0 + lrow] = acc[j][v];
  }
}

// ---------------------------------------------------------------------------
// Kernel 2: per-bt edge attention + aggregation + Out = Y @ W_out^T
// ---------------------------------------------------------------------------
#define Y_STRIDE 264

__global__ __launch_bounds__(256)
void k_edge_attn_out(const float* __restrict__ Xq, const float* __restrict__ Xv,
                     const float* __restrict__ avec, const float* __restrict__ A0,
                     const int* __restrict__ src, const int* __restrict__ dst,
                     const float* __restrict__ Wo, float* __restrict__ Out) {
  __shared__ float Ylds[Nn * Y_STRIDE];     // 67.6 KB
  __shared__ float esc[Ee * NHEAD];         // 8 KB (scores, then attn)
  __shared__ float segm[Nn * NHEAD];        // 1 KB
  __shared__ float segs[Nn * NHEAD];        // 1 KB
  __shared__ float al[Dd];                  // 1 KB
  __shared__ int   srcl[Ee], dstl[Ee];      // 4 KB

  const int tid = threadIdx.x;
  const int bt  = blockIdx.x;
  const size_t base = (size_t)bt * Nn * Dd;

  for (int i = tid; i < Nn * Y_STRIDE; i += 256) Ylds[i] = 0.f;
  for (int i = tid; i < Ee; i += 256) { srcl[i] = src[i]; dstl[i] = dst[i]; }
  al[tid] = avec[tid];                      // Dd == blockDim.x == 256
  __syncthreads();

  // ---- edge scores (2 edges / thread) ----
  for (int e = tid; e < Ee; e += 256) {
    const int se = srcl[e], de = dstl[e];
    const float bias = logf(A0[se * Nn + de] + EPSI);
    const float* qp = Xq + base + (size_t)se * Dd;
    const float* kp = Xq + base + (size_t)de * Dd;
#pragma unroll
    for (int h = 0; h < NHEAD; ++h) {
      float s = 0.f;
      for (int d = 0; d < DH; d += 4) {
        float4 q  = *reinterpret_cast<const float4*>(qp + h * DH + d);
        float4 k  = *reinterpret_cast<const float4*>(kp + h * DH + d);
        float4 av = *reinterpret_cast<const float4*>(&al[h * DH + d]);
        float x;
        x = q.x + k.x; s += (x > 0.f ? x : LRELU * x) * av.x;
        x = q.y + k.y; s += (x > 0.f ? x : LRELU * x) * av.y;
        x = q.z + k.z; s += (x > 0.f ? x : LRELU * x) * av.z;
        x = q.w + k.w; s += (x > 0.f ? x : LRELU * x) * av.w;
      }
      esc[e * NHEAD + h] = s + bias;
    }
  }
  __syncthreads();

  // ---- segment max + sum: one thread per (node, head) slot, deterministic ----
  {
    const int node = tid >> 2, h = tid & 3;
    float m = -3.402823466e38f;
    for (int e = 0; e < Ee; ++e)
      if (srcl[e] == node) m = fmaxf(m, esc[e * NHEAD + h]);
    float ssum = 0.f;
    for (int e = 0; e < Ee; ++e)
      if (srcl[e] == node) ssum += __expf(esc[e * NHEAD + h] - m);
    segm[tid] = m;
    segs[tid] = ssum;
  }
  __syncthreads();

  // ---- attn weights (in place) ----
  for (int e = tid; e < Ee; e += 256) {
    const int se = srcl[e];
#pragma unroll
    for (int h = 0; h < NHEAD; ++h)
      esc[e * NHEAD + h] =
          __expf(esc[e * NHEAD + h] - segm[se * NHEAD + h]) / segs[se * NHEAD + h];
  }
  __syncthreads();

  // ---- aggregation: each thread owns one output column -> no atomics ----
  {
    const int c = tid;          // column = h*DH + d
    const int h = c >> 6;
    const float* xvb = Xv + base;
    for (int e = 0; e < Ee; ++e) {
      const float w = esc[e * NHEAD + h];
      Ylds[srcl[e] * Y_STRIDE + c] += w * xvb[(size_t)dstl[e] * Dd + c];
    }
  }
  __syncthreads();

  // ---- Out[bt] (64x256) = Ylds @ Wo^T via WMMA ----
  const int wave  = tid >> 5;
  const int lane  = tid & 31;
  const int lrow  = lane & 15;
  const int khalf = (lane >> 4) << 1;
  const int rtile = wave >> 1;             // 0..3   (row tile of 16)
  const int cbase = (wave & 1) << 3;       // col tiles cbase..cbase+7

  v8f acc[8] = {};
  for (int k0 = 0; k0 < Dd; k0 += 4) {
    v2f a;
    {
      const float* ap = &Ylds[(rtile * 16 + lrow) * Y_STRIDE + k0 + khalf];
      a.x = ap[0]; a.y = ap[1];
    }
#pragma unroll
    for (int j = 0; j < 8; ++j) {
      int col = ((cbase + j) << 4) + lrow;
      const float* bp = Wo + (size_t)col * Dd + k0 + khalf;
      v2f b; b.x = bp[0]; b.y = bp[1];
      acc[j] = __builtin_amdgcn_wmma_f32_16x16x4_f32(false, a, false, b,
                                                     (short)0, acc[j], false, false);
    }
  }

  const int rbase = (lane >> 4) << 3;
#pragma unroll
  for (int j = 0; j < 8; ++j) {
    int n0 = (cbase + j) << 4;
#pragma unroll
    for (int v = 0; v < 8; ++v)
      Out[base + (size_t)(rtile * 16 + rbase + v) * Dd + n0 + lrow] = acc[j][v];
  }
}

// ---------------------------------------------------------------------------
extern "C" void kernel_launch(void* const* d_in, const int* in_sizes, int n_in,
                              void* d_out, int out_size, void* d_ws, size_t ws_size,
                              hipStream_t stream) {
  const float* H   = (const float*)d_in[0];
  const float* Wl  = (const float*)d_in[1];
  const float* Wv  = (const float*)d_in[2];
  const float* av  = (const float*)d_in[3];
  const float* Wo  = (const float*)d_in[4];
  const float* A0  = (const float*)d_in[5];
  const int*   src = (const int*)d_in[6];
  const int*   dst = (const int*)d_in[7];
  float* out = (float*)d_out;

  float* Xq = (float*)d_ws;                          // 64 MB
  float* Xv = Xq + (size_t)BT * Nn * Dd;             // 64 MB

  k_dual_gemm<<<(BT * Nn) / 32, 256, 0, stream>>>(H, Wl, Wv, Xq, Xv);
  k_edge_attn_out<<<BT, 256, 0, stream>>>(Xq, Xv, av, A0, src, dst, Wo, out);
}